// PositionalEncoding_85383949844653
// MI455X (gfx1250) — compile-verified
//
#include <hip/hip_runtime.h>

// Reference shapes (fixed by the harness):
//   x  : [32, 4096, 256] f32     pe : [5000, 256] f32
//   pos: [32, 4096] i32          lengths: [32] i32
//   out: [32, 4096, 256] f32
//
// out[b,t,:] = (t < lengths[b]) ? x[b,t,:] + pe[pos[b,t],:] : 0
//
// Pure bandwidth problem (~270 MB real HBM traffic @ 23.3 TB/s => ~12 us floor).
// Strategy:
//  - B128 coalesced accesses everywhere (64 consecutive lanes per 256-float row).
//  - Stage the streaming x operand into LDS via the CDNA5 async-copy path
//    (ASYNCcnt) with a non-temporal hint so the 128 MB x stream doesn't evict
//    the hot 5 MB pe table from L2; overlaps with the pe gather loads.
//  - pe gather is ALWAYS in-bounds (pos < 4096 < 5000), so no branch: issue
//    lengths/pos/pe loads unconditionally, apply the jagged mask only in the
//    final select. Removes EXEC divergence and two full loadcnt stalls.
//  - 2 vec4 per thread => 2 async ops + 2 independent gather chains per lane
//    for better latency hiding.
//  - Non-temporal B128 store for the write-once output.

typedef __attribute__((ext_vector_type(4))) float f4;

#define D4_SHIFT 6            // 64 float4 per row (D_MODEL = 256)
#define D4_MASK  63
#define L_SHIFT  12           // L = 4096
#define L_MASK   4095
#define BLOCK    256
#define VPT      2            // vec4 elements per thread

__global__ __launch_bounds__(BLOCK) void pe_jagged_add_kernel(
    const f4* __restrict__ x,        // [B*L*64] vec4
    const f4* __restrict__ pe,       // [5000*64] vec4
    const int* __restrict__ pos,     // [B*L]
    const int* __restrict__ lengths, // [B]
    f4* __restrict__ out)            // [B*L*64] vec4
{
    __shared__ f4 smem[BLOCK * VPT];

    const int tid  = threadIdx.x;
    const int base = blockIdx.x * (BLOCK * VPT);
    const int v0   = base + tid;          // first float4 index for this thread

    // ---- CDNA5 async copies: stream this block's 8 KB x tile into LDS ----
    // Per-lane 16B DMA, tracked by ASYNCcnt; issued under full EXEC.
    // NT hint: x is read exactly once -> evict-first near and far.
#pragma unroll
    for (int i = 0; i < VPT; ++i) {
        unsigned int       lds_off = (unsigned int)(size_t)&smem[i * BLOCK + tid];
        unsigned long long gaddr   = (unsigned long long)(const void*)(x + v0 + i * BLOCK);
        asm volatile("global_load_async_to_lds_b128 %0, %1, off th:TH_LOAD_NT"
                     :
                     : "v"(lds_off), "v"(gaddr)
                     : "memory");
    }

    // ---- index math (all shifts; block spans 8 whole rows of one batch) ----
    const int c    = v0 & D4_MASK;        // vec4 column, identical for both i
    const int row0 = v0 >> D4_SHIFT;      // b*L + t ; row_i = row0 + 4*i
    const int b    = row0 >> L_SHIFT;
    const int t0   = row0 & L_MASK;       // t_i = t0 + 4*i (no wrap in-block)

    const int len = lengths[b];           // one cacheline per block

    // pos loads: independent, issue together with lengths (no wait between).
    int pidx[VPT];
#pragma unroll
    for (int i = 0; i < VPT; ++i)
        pidx[i] = pos[row0 + 4 * i];      // always in [0, 4096) < MAX_LEN

    // pe gathers on the regular cached path (hot, L2-resident table).
    f4 pv[VPT];
#pragma unroll
    for (int i = 0; i < VPT; ++i)
        pv[i] = pe[(pidx[i] << D4_SHIFT) + c];

    // ---- consume the async tiles: wave-level wait, then LDS reads ----
    // Each thread reads only slots its own wave's async ops wrote, so
    // s_wait_asynccnt 0 suffices (no workgroup barrier needed).
    asm volatile("s_wait_asynccnt 0" ::: "memory");

#pragma unroll
    for (int i = 0; i < VPT; ++i) {
        const f4  xv    = smem[i * BLOCK + tid];
        const bool valid = (t0 + 4 * i) < len;
        const f4  s     = xv + pv[i];
        f4 o;
        o.x = valid ? s.x : 0.0f;
        o.y = valid ? s.y : 0.0f;
        o.z = valid ? s.z : 0.0f;
        o.w = valid ? s.w : 0.0f;
        // Write-once output: non-temporal store, don't pollute caches.
        __builtin_nontemporal_store(o, out + v0 + i * BLOCK);
    }
}

extern "C" void kernel_launch(void* const* d_in, const int* in_sizes, int n_in,
                              void* d_out, int out_size, void* d_ws, size_t ws_size,
                              hipStream_t stream) {
    const f4*  x       = (const f4*)d_in[0];
    const f4*  pe      = (const f4*)d_in[1];
    const int* pos     = (const int*)d_in[2];
    const int* lengths = (const int*)d_in[3];
    f4*        out     = (f4*)d_out;

    const int total4 = out_size / 4;              // 8,388,608 vec4 elements
    const int blocks = total4 / (BLOCK * VPT);    // 16,384 (exact, no tail)

    pe_jagged_add_kernel<<<blocks, BLOCK, 0, stream>>>(x, pe, pos, lengths, out);
}